// Wave3D_27676769255966
// MI455X (gfx1250) — compile-verified
//
#include <hip/hip_runtime.h>
#include <math.h>

typedef float v2f __attribute__((ext_vector_type(2)));
typedef float v8f __attribute__((ext_vector_type(8)));

#define PI_F 3.14159265358979323846f

// ---------------------------------------------------------------------------
// WMMA helper: D = A(16x4) * B(4x16) + C, fp32 end to end.
// A layout: lane L holds row m=L&15; k = 2*(L>>4) + component.
// B layout: lane L holds col n=L&15; k = 2*(L>>4) + component.
// C/D layout: lane L col n=L&15; VGPR r -> row r + 8*(L>>4).
// ---------------------------------------------------------------------------
__device__ __forceinline__ v8f wmma4(v2f a, v2f b, v8f c) {
  return __builtin_amdgcn_wmma_f32_16x16x4_f32(false, a, false, b, (short)0, c,
                                               false, false);
}

// ---------------------------------------------------------------------------
// Cos-matrix init: c64[m,k] = cos(m*(k+0.5)/64*pi)*sqrt(2/64), row0 * 1/sqrt2
// plus transposed copies; same for 16x16.
// ---------------------------------------------------------------------------
__global__ void cos_init_kernel(float* c64, float* c64T, float* c16, float* c16T) {
  int idx = blockIdx.x * blockDim.x + threadIdx.x;
  if (idx < 64 * 64) {
    int m = idx >> 6, k = idx & 63;
    float v = cosf((float)m * ((float)k + 0.5f) * (PI_F / 64.0f)) * sqrtf(2.0f / 64.0f);
    if (m == 0) v *= 0.70710678118654752f;
    c64[m * 64 + k] = v;
    c64T[k * 64 + m] = v;
  } else if (idx < 64 * 64 + 16 * 16) {
    int i2 = idx - 64 * 64;
    int m = i2 >> 4, k = i2 & 15;
    float v = cosf((float)m * ((float)k + 0.5f) * (PI_F / 16.0f)) * sqrtf(2.0f / 16.0f);
    if (m == 0) v *= 0.70710678118654752f;
    c16[m * 16 + k] = v;
    c16T[k * 16 + m] = v;
  }
}

// ---------------------------------------------------------------------------
// Depthwise 3x3x3 conv, input [B,C,T,H,W] channel-first, output channel-last
// yl[n=b*THW + t*HW + h*W + w, c].
// ---------------------------------------------------------------------------
__global__ void dwconv_kernel(const float* __restrict__ x,
                              const float* __restrict__ wgt,
                              const float* __restrict__ bias,
                              float* __restrict__ yl) {
  int tid = blockIdx.x * blockDim.x + threadIdx.x;
  if (tid >= 2 * 96 * 16 * 64 * 64) return;
  int w = tid & 63;
  int h = (tid >> 6) & 63;
  int t = (tid >> 12) & 15;
  int tmp = tid >> 16;     // b*96 + c
  int c = tmp % 96;
  int b = tmp / 96;
  const float* xb = x + (long)(b * 96 + c) * 65536;
  const float* wg = wgt + c * 27;
  float acc = 0.0f;
#pragma unroll
  for (int dt = -1; dt <= 1; ++dt) {
    int tt = t + dt;
    if (tt < 0 || tt >= 16) continue;
#pragma unroll
    for (int dh = -1; dh <= 1; ++dh) {
      int hh = h + dh;
      if (hh < 0 || hh >= 64) continue;
#pragma unroll
      for (int dw = -1; dw <= 1; ++dw) {
        int ww = w + dw;
        if (ww < 0 || ww >= 64) continue;
        acc += xb[tt * 4096 + hh * 64 + ww] * wg[(dt + 1) * 9 + (dh + 1) * 3 + (dw + 1)];
      }
    }
  }
  long n = (long)b * 65536 + t * 4096 + h * 64 + w;
  yl[n * 96 + c] = acc + bias[c];
}

// ---------------------------------------------------------------------------
// Channel GEMM, register-blocked over JB j-tiles per wave:
//   Out[row, j] = sum_k In[row, k] * Wm[j, k] + bias[j]
// Two base pointers only; every fragment is a base + compile-time-immediate
// global_load_b64, so the scheduler can keep many loads in flight.
// EPI: 0 = plain store; 1 = exact GELU; 2 = channel-first store with P=65536.
// ---------------------------------------------------------------------------
template <int K, int JB, int EPI>
__global__ __launch_bounds__(256) void gemm_kernel(
    const float* __restrict__ In, long in_rstride,
    const float* __restrict__ Wm, const float* __restrict__ bias,
    float* __restrict__ Out, long out_rstride, int Nrows, int J) {
  int wave = (blockIdx.x * blockDim.x + threadIdx.x) >> 5;
  int lane = threadIdx.x & 31;
  int jchunks = J / (16 * JB);
  long waves = (long)(Nrows >> 4) * jchunks;
  if (wave >= waves) return;
  int mt = wave / jchunks;
  int jc = wave % jchunks;
  int ln = lane & 15, half = lane >> 4;
  int m0 = mt << 4;
  int j0 = jc * 16 * JB;

  // Base pointers absorb all lane-dependent terms; offsets below are constants.
  const float* ap = In + (long)(m0 + ln) * in_rstride + 2 * half;
  const float* bp = Wm + (long)(j0 + ln) * K + 2 * half;

  v8f acc[JB];
#pragma unroll
  for (int jb = 0; jb < JB; ++jb) acc[jb] = (v8f){};

#pragma unroll
  for (int kk = 0; kk < K / 4; ++kk) {
    v2f a = *(const v2f*)(ap + kk * 4);
#pragma unroll
    for (int jb = 0; jb < JB; ++jb) {
      v2f b = *(const v2f*)(bp + jb * 16 * K + kk * 4);
      acc[jb] = wmma4(a, b, acc[jb]);
    }
  }
#pragma unroll
  for (int jb = 0; jb < JB; ++jb) {
    int j = j0 + jb * 16 + ln;
    float bj = bias[j];
#pragma unroll
    for (int r = 0; r < 8; ++r) {
      float v = acc[jb][r] + bj;
      long row = m0 + r + 8 * half;
      if (EPI == 1) v = 0.5f * v * (1.0f + erff(v * 0.70710678118654752f));
      if (EPI == 2) {
        long bb = row >> 16, p = row & 65535;   // P = THW = 65536
        Out[((bb * J + j) << 16) + p] = v;
      } else {
        Out[row * out_rstride + j] = v;
      }
    }
  }
}

// ---------------------------------------------------------------------------
// Generic axis transform, register-blocked over all MM/16 m-tiles per wave:
//   out[s][m, n] = sum_k mat[m,k] * in[s][k, n]
// A fragments: single base pointer + constant offsets into the tiny cos
// matrix (L0-resident). B fragments: two strided row pointers (runtime
// stride), strength-reduced over the fully unrolled k loop.
// in element  addr: in  + s*in_ss  + k*in_ks  + n
// out element addr: out + s*out_ss + m*out_ks + n
// ---------------------------------------------------------------------------
template <int MM, int KK>
__global__ __launch_bounds__(256) void transform_kernel(
    const float* __restrict__ mat, const float* __restrict__ in,
    float* __restrict__ out, int Ncols, long in_ss, long in_ks, long out_ss,
    long out_ks, int nslices) {
  constexpr int MT = MM >> 4;
  int wave = (blockIdx.x * blockDim.x + threadIdx.x) >> 5;
  int lane = threadIdx.x & 31;
  int ntiles = Ncols >> 4;
  long waves = (long)nslices * ntiles;
  if (wave >= waves) return;
  int s = wave / ntiles;
  int nt = wave % ntiles;
  int ln = lane & 15, half = lane >> 4;
  int n0 = nt << 4;

  const float* ap = mat + ln * KK + 2 * half;                  // constant offsets
  const float* bpx = in + (long)s * in_ss + (long)(2 * half) * in_ks + n0 + ln;
  const float* bpy = bpx + in_ks;
  long kstep = 4 * in_ks;
  float* outS = out + (long)s * out_ss;

  v8f acc[MT];
#pragma unroll
  for (int mt = 0; mt < MT; ++mt) acc[mt] = (v8f){};

#pragma unroll
  for (int kk = 0; kk < KK / 4; ++kk) {
    v2f b;
    b.x = bpx[(long)kk * kstep];
    b.y = bpy[(long)kk * kstep];
#pragma unroll
    for (int mt = 0; mt < MT; ++mt) {
      v2f a = *(const v2f*)(ap + mt * 16 * KK + kk * 4);
      acc[mt] = wmma4(a, b, acc[mt]);
    }
  }
  float* op = outS + n0 + ln;
#pragma unroll
  for (int mt = 0; mt < MT; ++mt) {
#pragma unroll
    for (int r = 0; r < 8; ++r)
      op[(long)(mt * 16 + r + 8 * half) * out_ks] = acc[mt][r];
  }
}

// ---------------------------------------------------------------------------
// Spectral elementwise:
// spec = exp(-a/2*tau) * ( cos(c*tau)*xu + sin(c*tau)/(c+1e-8)*(xv + a/2*xu) )
// xu/xv/spec are [n=b*65536+p, 96]; tau is [p, 96] broadcast over b.
// ---------------------------------------------------------------------------
__global__ void spec_kernel(const float* __restrict__ xu,
                            const float* __restrict__ xv,
                            const float* __restrict__ tau,
                            const float* __restrict__ cptr,
                            const float* __restrict__ aptr,
                            float* __restrict__ spec, int total) {
  int tid = blockIdx.x * blockDim.x + threadIdx.x;
  if (tid >= total) return;
  int c = tid % 96;
  int n = tid / 96;
  int p = n & 65535;
  float tv = tau[(long)p * 96 + c];
  float c0 = cptr[0], al = aptr[0];
  float ct = c0 * tv;
  float damp = __expf(-0.5f * al * tv);
  float u = xu[tid], v = xv[tid];
  spec[tid] = damp * (cosf(ct) * u + sinf(ct) / (c0 + 1e-8f) * (v + 0.5f * al * u));
}

// ---------------------------------------------------------------------------
// LayerNorm over C=96 + SiLU gate. One wave32 per row; 3 channels per lane.
// z comes from lin[row, 96 + c].
// ---------------------------------------------------------------------------
__global__ void ln_gate_kernel(const float* __restrict__ xf,
                               const float* __restrict__ lin,
                               const float* __restrict__ g,
                               const float* __restrict__ bt,
                               float* __restrict__ out, int rows) {
  int wave = (blockIdx.x * blockDim.x + threadIdx.x) >> 5;
  int lane = threadIdx.x & 31;
  if (wave >= rows) return;
  const float* xr = xf + (long)wave * 96;
  float v0 = xr[lane], v1 = xr[lane + 32], v2 = xr[lane + 64];
  float s = v0 + v1 + v2;
  float sq = v0 * v0 + v1 * v1 + v2 * v2;
#pragma unroll
  for (int off = 16; off >= 1; off >>= 1) {
    s += __shfl_xor(s, off, 32);
    sq += __shfl_xor(sq, off, 32);
  }
  float mu = s * (1.0f / 96.0f);
  float var = sq * (1.0f / 96.0f) - mu * mu;
  float rinv = rsqrtf(var + 1e-5f);
  const float* zr = lin + (long)wave * 192 + 96;
  float* orow = out + (long)wave * 96;
#pragma unroll
  for (int j = 0; j < 3; ++j) {
    int c = lane + j * 32;
    float xv = (j == 0) ? v0 : ((j == 1) ? v1 : v2);
    float nv = (xv - mu) * rinv * g[c] + bt[c];
    float z = zr[c];
    float silu = z / (1.0f + __expf(-z));
    orow[c] = nv * silu;
  }
}

// ---------------------------------------------------------------------------
// Host-side orchestration.
// ---------------------------------------------------------------------------
extern "C" void kernel_launch(void* const* d_in, const int* in_sizes, int n_in,
                              void* d_out, int out_size, void* d_ws, size_t ws_size,
                              hipStream_t stream) {
  const float* x          = (const float*)d_in[0];
  const float* freq_embed = (const float*)d_in[1];
  const float* dwconv_w   = (const float*)d_in[2];
  const float* dwconv_b   = (const float*)d_in[3];
  const float* linear_w   = (const float*)d_in[4];
  const float* linear_b   = (const float*)d_in[5];
  const float* v0_w       = (const float*)d_in[6];
  const float* v0_b       = (const float*)d_in[7];
  const float* tok_w      = (const float*)d_in[8];
  const float* tok_b      = (const float*)d_in[9];
  const float* c_s        = (const float*)d_in[10];
  const float* alpha_s    = (const float*)d_in[11];
  const float* ln_g       = (const float*)d_in[12];
  const float* ln_b       = (const float*)d_in[13];
  const float* out_w      = (const float*)d_in[14];
  const float* out_b      = (const float*)d_in[15];
  float* out = (float*)d_out;

  // Workspace layout (floats):
  const long NROWS = 131072;                 // B*T*H*W
  const long NC = NROWS * 96;                // 12,582,912
  float* ws   = (float*)d_ws;
  float* lin  = ws;                          // [131072, 192]
  float* RA   = lin + NROWS * 192;           // [131072, 96] (also yl)
  float* RC   = RA + NC;                     // [131072, 96]
  float* RF   = RC + NC;                     // [131072, 96]
  float* tau  = RF + NC;                     // [65536, 96]
  float* c64  = tau + 65536L * 96;           // 64x64
  float* c64T = c64 + 4096;
  float* c16  = c64T + 4096;                 // 16x16
  float* c16T = c16 + 256;

  const int TPB = 256;
  auto wblocks = [](long waves) { return (int)((waves + 7) / 8); };

  // 0) cos matrices
  cos_init_kernel<<<(4096 + 256 + TPB - 1) / TPB, TPB, 0, stream>>>(c64, c64T, c16, c16T);

  // 1) depthwise conv -> yl (RA)
  dwconv_kernel<<<(int)(NC / TPB), TPB, 0, stream>>>(x, dwconv_w, dwconv_b, RA);

  // 2) lin = yl @ linear_w.T + b   [131072, 192]   (2 j-chunks of 96)
  gemm_kernel<96, 6, 0><<<wblocks((NROWS / 16) * 2), TPB, 0, stream>>>(
      RA, 96, linear_w, linear_b, lin, 192, (int)NROWS, 192);

  // 3) v0 = xg @ v0_w.T + b  (xg = lin[:, :96], row stride 192) -> RC
  gemm_kernel<96, 6, 0><<<wblocks(NROWS / 16), TPB, 0, stream>>>(
      lin, 192, v0_w, v0_b, RC, 96, (int)NROWS, 96);

  // 4) tau = GELU(freq_embed @ tok_w.T + b) -> tau
  gemm_kernel<96, 6, 1><<<wblocks(65536 / 16), TPB, 0, stream>>>(
      freq_embed, 96, tok_w, tok_b, tau, 96, 65536, 96);

  // 5) DCT3 of xg: W (lin->RA), H (RA->RF), T (RF->RA)
  transform_kernel<64, 64><<<wblocks(2048L * 6), TPB, 0, stream>>>(
      c64, lin, RA, 96, 12288, 192, 6144, 96, 2048);
  transform_kernel<64, 64><<<wblocks(32L * 384), TPB, 0, stream>>>(
      c64, RA, RF, 6144, 393216, 6144, 393216, 6144, 32);
  transform_kernel<16, 16><<<wblocks(2L * 24576), TPB, 0, stream>>>(
      c16, RF, RA, 393216, 6291456, 393216, 6291456, 393216, 2);

  // 6) DCT3 of v0: W (RC->RF), H (RF->RC), T (RC->RF)
  transform_kernel<64, 64><<<wblocks(2048L * 6), TPB, 0, stream>>>(
      c64, RC, RF, 96, 6144, 96, 6144, 96, 2048);
  transform_kernel<64, 64><<<wblocks(32L * 384), TPB, 0, stream>>>(
      c64, RF, RC, 6144, 393216, 6144, 393216, 6144, 32);
  transform_kernel<16, 16><<<wblocks(2L * 24576), TPB, 0, stream>>>(
      c16, RC, RF, 393216, 6291456, 393216, 6291456, 393216, 2);

  // 7) spectral combine: xu=RA, xv=RF -> spec in RC
  spec_kernel<<<(int)(NC / TPB), TPB, 0, stream>>>(RA, RF, tau, c_s, alpha_s, RC, (int)NC);

  // 8) iDCT3 (transposed matrices): W (RC->RA), H (RA->RC), T (RC->RA)
  transform_kernel<64, 64><<<wblocks(2048L * 6), TPB, 0, stream>>>(
      c64T, RC, RA, 96, 6144, 96, 6144, 96, 2048);
  transform_kernel<64, 64><<<wblocks(32L * 384), TPB, 0, stream>>>(
      c64T, RA, RC, 6144, 393216, 6144, 393216, 6144, 32);
  transform_kernel<16, 16><<<wblocks(2L * 24576), TPB, 0, stream>>>(
      c16T, RC, RA, 393216, 6291456, 393216, 6291456, 393216, 2);

  // 9) LayerNorm + SiLU gate: xf=RA, z from lin -> RC
  ln_gate_kernel<<<wblocks(NROWS), TPB, 0, stream>>>(RA, lin, ln_g, ln_b, RC, (int)NROWS);

  // 10) out = gated @ out_w.T + b, stored channel-first [B,C,T,H,W]
  gemm_kernel<96, 6, 2><<<wblocks(NROWS / 16), TPB, 0, stream>>>(
      RC, 96, out_w, out_b, out, 0, (int)NROWS, 96);
}